// Attn_34754875359623
// MI455X (gfx1250) — compile-verified
//
#include <hip/hip_runtime.h>
#include <hip/hip_bf16.h>

typedef __attribute__((ext_vector_type(2))) float v2f;
typedef __attribute__((ext_vector_type(8))) float v8f;

#define B_   32
#define S_   4096
#define H_   256

// ---------------------------------------------------------------------------
// Kernel 1: prep.  blocks 0..31 -> c[b] and ||v_b||^2 ; block 32 -> wu_eff[h]
// wsmall layout (floats): [0..255]=wu_eff, [256..287]=c[b], [288..319]=vnorm2[b]
// ---------------------------------------------------------------------------
__global__ void __launch_bounds__(256)
attn_prep(const float* __restrict__ v, const float* __restrict__ W,
          const float* __restrict__ b_attn, const float* __restrict__ v_param,
          float* __restrict__ wsmall) {
    __shared__ float red[256];
    const int tid = threadIdx.x;
    if (blockIdx.x == 32) {
        // wu_eff[h] = sum_k v_param[k] * W[k, h]   (W row stride = 2H)
        float acc = 0.f;
        for (int k = 0; k < H_; ++k)
            acc = fmaf(v_param[k], W[k * (2 * H_) + tid], acc);
        wsmall[tid] = acc;
        return;
    }
    const int b = blockIdx.x;
    const float* vb = v + b * H_;
    // thread tid owns k=tid:  t = v_param[k] * (b_attn[k] + sum_h v[b,h]*W[k, H+h])
    float t = b_attn[tid];
    const float* wrow = W + tid * (2 * H_) + H_;
    for (int h = 0; h < H_; ++h) t = fmaf(vb[h], wrow[h], t);
    t *= v_param[tid];
    red[tid] = t; __syncthreads();
    for (int s = 128; s > 0; s >>= 1) {
        if (tid < s) red[tid] += red[tid + s];
        __syncthreads();
    }
    if (tid == 0) wsmall[256 + b] = red[0];
    __syncthreads();
    float q = vb[tid];
    red[tid] = q * q; __syncthreads();
    for (int s = 128; s > 0; s >>= 1) {
        if (tid < s) red[tid] += red[tid + s];
        __syncthreads();
    }
    if (tid == 0) wsmall[288 + b] = red[0];
}

// ---------------------------------------------------------------------------
// Kernel 2: main streaming pass.  One wave = one 16-row tile of u (16x256).
// A (16x4 per step): row0 = wu_eff chunk, row1 = v_b chunk, rows 2..15 = 0.
// B (4x16 per step): the u tile chunk.  D col-layout puts score in acc[0],
// u.v in acc[1], per lane n = row n.  ||u||^2 via per-lane VALU + shfl_xor(16).
// ---------------------------------------------------------------------------
__global__ void __launch_bounds__(256)
attn_stream(const float* __restrict__ u, const float* __restrict__ wsmall,
            const float* __restrict__ v, float* __restrict__ scores,
            float* __restrict__ dist) {
    __shared__ float s_A[3][H_];            // [0]=wu_eff, [1]=v_b, [2]=zeros
    const int tid = threadIdx.x;
    const int b = blockIdx.x >> 5;          // 32 blocks per batch element
    s_A[0][tid] = wsmall[tid];
    s_A[1][tid] = v[b * H_ + tid];
    s_A[2][tid] = 0.f;
    __syncthreads();

    const int warp = tid >> 5;
    const int lane = tid & 31;
    const int tile = (blockIdx.x << 3) + warp;     // [0, 8192)
    const int m    = lane & 15;
    const int half = lane >> 4;

    const float* up = u + ((size_t)(tile << 4) + (size_t)m) * H_ + 2 * half;
    const float* ap = &s_A[m < 2 ? m : 2][2 * half];

    v8f acc = {0.f, 0.f, 0.f, 0.f, 0.f, 0.f, 0.f, 0.f};
    float sumsq = 0.f;
#pragma unroll
    for (int kk = 0; kk < 64; ++kk) {
        v2f ub = *(const v2f*)(up + 4 * kk);   // B fragment (u tile)
        v2f aa = *(const v2f*)(ap + 4 * kk);   // A fragment (weights)
        sumsq = fmaf(ub.x, ub.x, sumsq);
        sumsq = fmaf(ub.y, ub.y, sumsq);
        acc = __builtin_amdgcn_wmma_f32_16x16x4_f32(
            /*neg_a=*/false, aa, /*neg_b=*/false, ub,
            /*c_mod=*/(short)0, acc, /*reuse_a=*/false, /*reuse_b=*/false);
    }
    // lane L and lane L^16 cover complementary K-halves of row (L&15)
    float tot = sumsq + __shfl_xor(sumsq, 16, 32);

    if (lane < 16) {
        const float cb  = wsmall[256 + b];
        const float vn2 = wsmall[288 + b];
        const int idx = (tile << 4) + lane;        // == b*S + s0 + lane
        scores[idx] = acc[0] + cb;                 // raw score
        float d2 = tot - 2.f * acc[1] + vn2;       // ||u||^2 - 2 u.v + ||v||^2
        dist[idx] = sqrtf(fmaxf(d2, 0.f));
    }
}

// ---------------------------------------------------------------------------
// Kernel 3: per-b softmax over S=4096 + weighted distance.  One block per b.
// Writes attn in-place over the raw scores (reads complete before writes).
// ---------------------------------------------------------------------------
__global__ void __launch_bounds__(256)
attn_finalize(const float* __restrict__ scores, const float* __restrict__ dist,
              float* __restrict__ out_wd, float* __restrict__ out_attn) {
    __shared__ float red[256];
    __shared__ float e[S_];
    const int b = blockIdx.x;
    const int tid = threadIdx.x;
    const float* sc = scores + b * S_;
    const float* dd = dist + b * S_;

    float mx = -3.402823466e38f;
    for (int i = tid; i < S_; i += 256) mx = fmaxf(mx, sc[i]);
    red[tid] = mx; __syncthreads();
    for (int s = 128; s > 0; s >>= 1) {
        if (tid < s) red[tid] = fmaxf(red[tid], red[tid + s]);
        __syncthreads();
    }
    mx = red[0]; __syncthreads();

    float se = 0.f, sed = 0.f;
    for (int i = tid; i < S_; i += 256) {
        float ev = __expf(sc[i] - mx);
        e[i] = ev;
        se += ev;
        sed = fmaf(ev, dd[i], sed);
    }
    red[tid] = se; __syncthreads();
    for (int s = 128; s > 0; s >>= 1) {
        if (tid < s) red[tid] += red[tid + s];
        __syncthreads();
    }
    se = red[0]; __syncthreads();
    red[tid] = sed; __syncthreads();
    for (int s = 128; s > 0; s >>= 1) {
        if (tid < s) red[tid] += red[tid + s];
        __syncthreads();
    }
    sed = red[0]; __syncthreads();

    if (tid == 0) out_wd[b] = sed / se;
    const float inv = 1.f / se;
    for (int i = tid; i < S_; i += 256)
        out_attn[b * S_ + i] = e[i] * inv;
}

// ---------------------------------------------------------------------------
extern "C" void kernel_launch(void* const* d_in, const int* in_sizes, int n_in,
                              void* d_out, int out_size, void* d_ws, size_t ws_size,
                              hipStream_t stream) {
    const float* u  = (const float*)d_in[0];   // (32, 4096, 256)
    const float* v  = (const float*)d_in[1];   // (32, 256)
    const float* W  = (const float*)d_in[2];   // (256, 512)
    const float* ba = (const float*)d_in[3];   // (256,)
    const float* vp = (const float*)d_in[4];   // (256,)

    float* out    = (float*)d_out;             // [0..32) = w_d, [32..) = attn
    float* wsmall = (float*)d_ws;              // 512 floats of prep data
    float* dist   = wsmall + 512;              // 32*4096 floats
    float* scores = out + B_;                  // reuse attn slots for raw scores

    attn_prep<<<33, 256, 0, stream>>>(v, W, ba, vp, wsmall);
    attn_stream<<<(B_ * S_) / 128, 256, 0, stream>>>(u, wsmall, v, scores, dist);
    attn_finalize<<<B_, 256, 0, stream>>>(scores, dist, out, out + B_);
}